// Decoder_5927054868853
// MI455X (gfx1250) — compile-verified
//
#include <hip/hip_runtime.h>
#include <math.h>

// Shapes: B=16, T=64, S=128, D_ENC=512, U=256, V=32000, X=D+U=768, G=4U=1024
#define B_  16
#define T_  64
#define S_  128
#define D_  512
#define U_  256
#define V_  32000
#define X_  768
#define G_  1024

typedef float v2f __attribute__((ext_vector_type(2)));
typedef float v8f __attribute__((ext_vector_type(8)));

// ---------------------------------------------------------------------------
// keys = enc_outputs.view(2048,512) @ W2(512,256) + b2   (one-time)
// One 16x16 tile per wave, fp32 WMMA K=4 steps.
// ---------------------------------------------------------------------------
__global__ void keys_kernel(const float* __restrict__ enc, const float* __restrict__ W2,
                            const float* __restrict__ b2, float* __restrict__ keys) {
    const int lane = threadIdx.x & 31;
    const int wv   = threadIdx.x >> 5;
    const int tile = blockIdx.x * 4 + wv;          // 0..2047
    const int tm   = tile >> 4;                    // 128 row tiles (2048 rows)
    const int tn   = tile & 15;                    // 16 col tiles (256 cols)
    const int half = lane >> 4;
    const int l16  = lane & 15;
    const int row  = tm * 16 + l16;                // A row held by this lane
    const int n    = tn * 16 + l16;                // B column held by this lane
    v8f acc = {};
    for (int k = 0; k < D_; k += 4) {
        const int ka = k + 2 * half;               // lanes 16-31 hold K pair {2,3}
        v2f a, b;
        a.x = enc[row * D_ + ka];
        a.y = enc[row * D_ + ka + 1];
        b.x = W2[(ka    ) * U_ + n];
        b.y = W2[(ka + 1) * U_ + n];
        acc = __builtin_amdgcn_wmma_f32_16x16x4_f32(false, a, false, b, (short)0, acc,
                                                    false, false);
    }
    const float bias = b2[n];
#pragma unroll
    for (int r = 0; r < 8; ++r) {                  // VGPR r -> row r + 8*half
        const int m = tm * 16 + r + 8 * half;
        keys[m * U_ + n] = acc[r] + bias;
    }
}

// ---------------------------------------------------------------------------
// Bahdanau attention + context + concat with embedded token. One block per b.
// ---------------------------------------------------------------------------
__global__ void attn_kernel(const float* __restrict__ h, const float* __restrict__ keys,
                            const float* __restrict__ enc, const float* __restrict__ W1,
                            const float* __restrict__ b1, const float* __restrict__ Va,
                            const float* __restrict__ bV, const float* __restrict__ emb,
                            const int* __restrict__ dec, float* __restrict__ x, int t) {
    const int b   = blockIdx.x;
    const int tid = threadIdx.x;                   // 256 threads
    __shared__ float q[U_];
    __shared__ float sc[S_];
    __shared__ float red[256];

    // q = h[b] @ W1 + b1 (coalesced over output column)
    {
        float acc = b1[tid];
        const float* hb = h + b * U_;
        for (int k = 0; k < U_; ++k) acc += hb[k] * W1[k * U_ + tid];
        q[tid] = acc;
    }
    __syncthreads();

    // score[s] = tanh(q + keys[b,s,:]) . Va + bV   (2 threads per s)
    {
        const int s   = tid >> 1;
        const int hlf = tid & 1;
        const float* kr = keys + (b * S_ + s) * U_ + hlf * 128;
        const float* qv = q + hlf * 128;
        const float* va = Va + hlf * 128;
        float p = 0.f;
        for (int k = 0; k < 128; ++k) p += tanhf(qv[k] + kr[k]) * va[k];
        red[tid] = p;
    }
    __syncthreads();
    if ((tid & 1) == 0) sc[tid >> 1] = red[tid] + red[tid + 1] + bV[0];
    __syncthreads();

    // softmax over S=128
    if (tid < S_) red[tid] = sc[tid];
    __syncthreads();
    for (int w = 64; w > 0; w >>= 1) {
        if (tid < w) red[tid] = fmaxf(red[tid], red[tid + w]);
        __syncthreads();
    }
    const float mx = red[0];
    __syncthreads();
    if (tid < S_) { const float e = expf(sc[tid] - mx); sc[tid] = e; red[tid] = e; }
    __syncthreads();
    for (int w = 64; w > 0; w >>= 1) {
        if (tid < w) red[tid] += red[tid + w];
        __syncthreads();
    }
    const float inv = 1.f / red[0];
    __syncthreads();
    if (tid < S_) sc[tid] *= inv;
    __syncthreads();

    // ctx = sum_s attn[s] * enc[b,s,:]; x = [ctx(512) | emb[tok](256)]
    float c0 = 0.f, c1 = 0.f;
    const float* eb = enc + (size_t)b * S_ * D_;
    for (int s = 0; s < S_; ++s) {
        const float a = sc[s];
        c0 += a * eb[s * D_ + tid];
        c1 += a * eb[s * D_ + 256 + tid];
    }
    float* xb = x + b * X_;
    xb[tid]       = c0;
    xb[256 + tid] = c1;
    const int tk = dec[b * T_ + t];
    xb[512 + tid] = emb[tk * U_ + tid];
}

// ---------------------------------------------------------------------------
// z = x(16x768) @ Wx(768x1024) + h(16x256) @ Wh(256x1024) + b_lstm
// ---------------------------------------------------------------------------
__global__ void lstm_gemm_kernel(const float* __restrict__ x, const float* __restrict__ h,
                                 const float* __restrict__ Wx, const float* __restrict__ Wh,
                                 const float* __restrict__ bl, float* __restrict__ z) {
    const int lane = threadIdx.x & 31;
    const int wv   = threadIdx.x >> 5;
    const int tn   = blockIdx.x * 4 + wv;          // 0..63
    const int half = lane >> 4;
    const int l16  = lane & 15;
    const int n    = tn * 16 + l16;
    v8f acc = {};
    for (int k = 0; k < X_; k += 4) {
        const int ka = k + 2 * half;
        v2f a, b;
        a.x = x[l16 * X_ + ka];
        a.y = x[l16 * X_ + ka + 1];
        b.x = Wx[(ka    ) * G_ + n];
        b.y = Wx[(ka + 1) * G_ + n];
        acc = __builtin_amdgcn_wmma_f32_16x16x4_f32(false, a, false, b, (short)0, acc,
                                                    false, false);
    }
    for (int k = 0; k < U_; k += 4) {
        const int ka = k + 2 * half;
        v2f a, b;
        a.x = h[l16 * U_ + ka];
        a.y = h[l16 * U_ + ka + 1];
        b.x = Wh[(ka    ) * G_ + n];
        b.y = Wh[(ka + 1) * G_ + n];
        acc = __builtin_amdgcn_wmma_f32_16x16x4_f32(false, a, false, b, (short)0, acc,
                                                    false, false);
    }
    const float bias = bl[n];
#pragma unroll
    for (int r = 0; r < 8; ++r) {
        const int m = r + 8 * half;                // batch row 0..15
        z[m * G_ + n] = acc[r] + bias;
    }
}

// ---------------------------------------------------------------------------
// LSTM gates (Keras order i,f,g,o): update c,h in place.
// ---------------------------------------------------------------------------
__global__ void gates_kernel(const float* __restrict__ z, float* __restrict__ c,
                             float* __restrict__ h) {
    const int idx = blockIdx.x * blockDim.x + threadIdx.x;   // 0..4095
    const int b = idx >> 8, u = idx & 255;
    const float* zb = z + b * G_;
    const float zi = zb[u], zf = zb[256 + u], zg = zb[512 + u], zo = zb[768 + u];
    const float si = 1.f / (1.f + expf(-zi));
    const float sf = 1.f / (1.f + expf(-zf));
    const float so = 1.f / (1.f + expf(-zo));
    const float cn = sf * c[idx] + si * tanhf(zg);
    c[idx] = cn;
    h[idx] = so * tanhf(cn);
}

// ---------------------------------------------------------------------------
// logits[b, t, :] = h(16x256) @ Wd(256x32000) + bd  -- dominant GEMM.
// 2000 N-tiles, one per wave; Wd streams from L2 (32.7MB < 192MB L2).
// ---------------------------------------------------------------------------
__global__ void proj_kernel(const float* __restrict__ h, const float* __restrict__ Wd,
                            const float* __restrict__ bd, float* __restrict__ out, int t) {
    const int lane = threadIdx.x & 31;
    const int wv   = threadIdx.x >> 5;
    const int tn   = blockIdx.x * 4 + wv;          // 0..1999
    const int half = lane >> 4;
    const int l16  = lane & 15;
    const int n    = tn * 16 + l16;
    v8f acc = {};
    for (int k = 0; k < U_; k += 4) {
        const int ka = k + 2 * half;
        v2f a, b;
        a.x = h[l16 * U_ + ka];
        a.y = h[l16 * U_ + ka + 1];
        b.x = Wd[(size_t)(ka    ) * V_ + n];
        b.y = Wd[(size_t)(ka + 1) * V_ + n];
        if (ka + 18 < U_)   // pull upcoming Wd rows toward the WGP (global_prefetch_b8)
            __builtin_prefetch(&Wd[(size_t)(ka + 16) * V_ + n], 0, 1);
        acc = __builtin_amdgcn_wmma_f32_16x16x4_f32(false, a, false, b, (short)0, acc,
                                                    false, false);
    }
    const float bias = bd[n];
#pragma unroll
    for (int r = 0; r < 8; ++r) {
        const int m = r + 8 * half;
        out[(size_t)m * T_ * V_ + (size_t)t * V_ + n] = acc[r] + bias;
    }
}

// ---------------------------------------------------------------------------
// Row softmax over V=32000, in place on d_out[b, t, :]. One block per b.
// ---------------------------------------------------------------------------
__global__ void softmax_kernel(float* __restrict__ out, int t) {
    const int b = blockIdx.x;
    float* row = out + (size_t)b * T_ * V_ + (size_t)t * V_;
    __shared__ float red[1024];
    const int tid = threadIdx.x;
    float m = -INFINITY;
    for (int i = tid; i < V_; i += 1024) m = fmaxf(m, row[i]);
    red[tid] = m;
    __syncthreads();
    for (int w = 512; w > 0; w >>= 1) {
        if (tid < w) red[tid] = fmaxf(red[tid], red[tid + w]);
        __syncthreads();
    }
    m = red[0];
    __syncthreads();
    float sum = 0.f;
    for (int i = tid; i < V_; i += 1024) {
        const float e = expf(row[i] - m);
        row[i] = e;
        sum += e;
    }
    red[tid] = sum;
    __syncthreads();
    for (int w = 512; w > 0; w >>= 1) {
        if (tid < w) red[tid] += red[tid + w];
        __syncthreads();
    }
    const float inv = 1.f / red[0];
    for (int i = tid; i < V_; i += 1024) row[i] *= inv;
}

// ---------------------------------------------------------------------------
extern "C" void kernel_launch(void* const* d_in, const int* in_sizes, int n_in,
                              void* d_out, int out_size, void* d_ws, size_t ws_size,
                              hipStream_t stream) {
    const int*   dec = (const int*)  d_in[0];
    const float* enc = (const float*)d_in[1];
    const float* h0  = (const float*)d_in[2];
    const float* c0  = (const float*)d_in[3];
    const float* emb = (const float*)d_in[4];
    const float* W1  = (const float*)d_in[5];
    const float* b1  = (const float*)d_in[6];
    const float* W2  = (const float*)d_in[7];
    const float* b2  = (const float*)d_in[8];
    const float* Va  = (const float*)d_in[9];
    const float* bV  = (const float*)d_in[10];
    const float* Wx  = (const float*)d_in[11];
    const float* Wh  = (const float*)d_in[12];
    const float* bl  = (const float*)d_in[13];
    const float* Wd  = (const float*)d_in[14];
    const float* bd  = (const float*)d_in[15];
    float* out = (float*)d_out;

    float* ws   = (float*)d_ws;
    float* keys = ws;                       // 2048*256
    float* hbuf = keys + 2048 * U_;         // 16*256
    float* cbuf = hbuf + B_ * U_;           // 16*256
    float* xbuf = cbuf + B_ * U_;           // 16*768
    float* zbuf = xbuf + B_ * X_;           // 16*1024

    hipMemcpyAsync(hbuf, h0, B_ * U_ * sizeof(float), hipMemcpyDeviceToDevice, stream);
    hipMemcpyAsync(cbuf, c0, B_ * U_ * sizeof(float), hipMemcpyDeviceToDevice, stream);

    keys_kernel<<<512, 128, 0, stream>>>(enc, W2, b2, keys);   // 2048 tiles / 4 waves

    for (int t = 0; t < T_; ++t) {
        attn_kernel<<<B_, 256, 0, stream>>>(hbuf, keys, enc, W1, b1, Va, bV, emb, dec,
                                            xbuf, t);
        lstm_gemm_kernel<<<16, 128, 0, stream>>>(xbuf, hbuf, Wx, Wh, bl, zbuf);
        gates_kernel<<<16, 256, 0, stream>>>(zbuf, cbuf, hbuf);
        proj_kernel<<<500, 128, 0, stream>>>(hbuf, Wd, bd, out, t);  // 2000 tiles
        softmax_kernel<<<B_, 1024, 0, stream>>>(out, t);
    }
}